// VideoEmbedding_81363860455848
// MI455X (gfx1250) — compile-verified
//
#include <hip/hip_runtime.h>
#include <math.h>

#define NF 6
#define KDIM 13
#define KP 17        // padded K stride in LDS
#define DDIM 64
#define VMAX 128     // bins in LDS histograms (reference has 128 videos)

typedef __attribute__((ext_vector_type(2))) float v2f;
typedef __attribute__((ext_vector_type(8))) float v8f;

// workspace layout (ints):
//   [0   ..127] counts
//   [128 ..255] exclusive offsets
//   [256 ..383] scatter cursors
//   [512 ..   ] bucketed sample indices (N ints)

__global__ void ve_zero_ws(int* ws) {
    int i = threadIdx.x;
    if (i < 384) ws[i] = 0;
}

// Hierarchical histogram: LDS bins per block, then <=128 global atomics/block.
__global__ __launch_bounds__(1024) void ve_count(const int* __restrict__ vid,
                                                 int* __restrict__ ws, int n) {
    __shared__ int h[VMAX];
    if (threadIdx.x < VMAX) h[threadIdx.x] = 0;
    __syncthreads();
    int i = blockIdx.x * blockDim.x + threadIdx.x;
    if (i < n) atomicAdd(&h[vid[i]], 1);
    __syncthreads();
    if (threadIdx.x < VMAX) {
        int c = h[threadIdx.x];
        if (c) atomicAdd(&ws[threadIdx.x], c);
    }
}

__global__ void ve_scan(int* ws, int V) {
    if (threadIdx.x == 0) {
        int acc = 0;
        for (int v = 0; v < V; ++v) { ws[128 + v] = acc; acc += ws[v]; }
    }
}

// Hierarchical scatter: LDS histogram assigns in-block ranks; one global atomic
// per (block, video) reserves a contiguous bucket range; writes are atomic-free.
__global__ __launch_bounds__(1024) void ve_scatter(const int* __restrict__ vid,
                                                   int* __restrict__ ws, int n) {
    __shared__ int h[VMAX];     // in-block counts -> ranks
    __shared__ int basei[VMAX]; // offsets[v] + reserved cursor for this block
    if (threadIdx.x < VMAX) h[threadIdx.x] = 0;
    __syncthreads();

    int i = blockIdx.x * blockDim.x + threadIdx.x;
    int v = 0, rank = 0;
    const bool valid = i < n;
    if (valid) {
        v = vid[i];
        rank = atomicAdd(&h[v], 1);
    }
    __syncthreads();

    if (threadIdx.x < VMAX) {
        int c = h[threadIdx.x];
        int cur = c ? atomicAdd(&ws[256 + threadIdx.x], c) : 0;
        basei[threadIdx.x] = ws[128 + threadIdx.x] + cur;
    }
    __syncthreads();

    if (valid) ws[512 + basei[v] + rank] = i;
}

// Branchless Fourier basis element:
//   k==0 -> 1 ; 1..6 -> sin(2^(k-1) pi t) ; 7..12 -> cos(2^(k-7) pi t) ; >=13 -> 0
__device__ __forceinline__ float basis_elem(float pt, int k) {
    const bool isCos = k >= 7;
    const int j = isCos ? (k - 7) : (k - 1);
    float angle = __builtin_amdgcn_ldexpf(pt, j);
    angle += isCos ? 1.57079632679489661923f : 0.0f;   // cos(x) = sin(x + pi/2)
    float v = __sinf(angle);
    v = (k == 0) ? 1.0f : v;
    v = (k >= KDIM) ? 0.0f : v;
    return v;
}

// One block per (video, tile-slice); 8 waves/block; each wave owns 16-sample tiles.
// Tile GEMM: C[16x64] = A[16x16(basis,padded)] x B[16x64] via 4x4 wmma_f32_16x16x4.
__global__ __launch_bounds__(256) void ve_wmma(
    const float* __restrict__ times,
    const float* __restrict__ W,       // [V][64][13]
    const int*   __restrict__ ws,
    float*       __restrict__ out)     // [N][64]
{
    __shared__ float Wlds[DDIM * KP];

    const int v      = blockIdx.x;
    const int cnt    = ws[v];
    const int bstart = ws[128 + v];
    const int* bucket = ws + 512;

    // Stage this video's weights into LDS as Wlds[d*KP + k], zero-padded k>=13.
    for (int e = threadIdx.x; e < DDIM * KP; e += blockDim.x) {
        int d = e / KP, k = e % KP;
        Wlds[e] = (k < KDIM) ? W[(v * DDIM + d) * KDIM + k] : 0.0f;
    }
    __syncthreads();

    const int lane  = threadIdx.x & 31;
    // Scalar wave id -> tile-loop control, `remain`, `full` live in SGPRs.
    const int wave  = __builtin_amdgcn_readfirstlane(threadIdx.x >> 5);
    const int lo    = lane & 15;         // A-row / B-column within half-wave
    const int hi2   = (lane >> 4) * 2;   // K sub-offset inside a K=4 chunk
    const int mbase = (lane >> 4) * 8;   // first C row this lane stores

    // Hoist B fragments (loop-invariant): bfrag[db][c] covers dims db*16+lo, K=c*4+hi2+{0,1}
    v2f bfrag[4][4];
    #pragma unroll
    for (int db = 0; db < 4; ++db) {
        const float* wrow = &Wlds[(db * 16 + lo) * KP];
        #pragma unroll
        for (int c = 0; c < 4; ++c) {
            bfrag[db][c][0] = wrow[c * 4 + hi2];
            bfrag[db][c][1] = wrow[c * 4 + hi2 + 1];
        }
    }

    const int wavesPerVideo = gridDim.y * 8;
    const int numTiles = (cnt + 15) >> 4;

    for (int tile = blockIdx.y * 8 + wave; tile < numTiles; tile += wavesPerVideo) {
        const int base   = tile << 4;          // scalar
        const int remain = cnt - base;         // scalar, >= 1
        const bool full  = remain >= 16;       // scalar -> s_cbranch

        // Clamped (always in-bounds) sample index for this lane's row `lo`;
        // garbage basis on invalid rows is fine: those C rows are never stored.
        int clampRow = base + lo; clampRow = clampRow < cnt - 1 ? clampRow : cnt - 1;
        const int sidxRaw = bucket[bstart + clampRow];
        const float t  = times[sidxRaw];
        const float pt = t * 3.14159265358979323846f;

        // A fragments: 16x4 f32 layout (lanes 0-15: K=hi2,hi2+1 ; lanes 16-31: K=hi2+2..)
        v2f a[4];
        #pragma unroll
        for (int c = 0; c < 4; ++c) {
            a[c][0] = basis_elem(pt, c * 4 + hi2);
            a[c][1] = basis_elem(pt, c * 4 + hi2 + 1);
        }

        // Output row base offsets: lanes 0-15 hold rows 0-15's sample idx -> shuffle.
        int rbase[8];
        #pragma unroll
        for (int r = 0; r < 8; ++r)
            rbase[r] = __shfl(sidxRaw, mbase + r) * DDIM + lo;

        if (full) {
            #pragma unroll
            for (int db = 0; db < 4; ++db) {
                v8f acc = {};
                #pragma unroll
                for (int c = 0; c < 4; ++c) {
                    acc = __builtin_amdgcn_wmma_f32_16x16x4_f32(
                            false, a[c], false, bfrag[db][c],
                            (short)0, acc, false, false);
                }
                #pragma unroll
                for (int r = 0; r < 8; ++r)
                    __builtin_nontemporal_store(acc[r], &out[rbase[r] + db * 16]);
            }
        } else {
            #pragma unroll
            for (int db = 0; db < 4; ++db) {
                v8f acc = {};
                #pragma unroll
                for (int c = 0; c < 4; ++c) {
                    acc = __builtin_amdgcn_wmma_f32_16x16x4_f32(
                            false, a[c], false, bfrag[db][c],
                            (short)0, acc, false, false);
                }
                #pragma unroll
                for (int r = 0; r < 8; ++r)
                    if (mbase + r < remain)
                        __builtin_nontemporal_store(acc[r], &out[rbase[r] + db * 16]);
            }
        }
    }
}

extern "C" void kernel_launch(void* const* d_in, const int* in_sizes, int n_in,
                              void* d_out, int out_size, void* d_ws, size_t ws_size,
                              hipStream_t stream) {
    const float* times = (const float*)d_in[0];
    const int*   vids  = (const int*)d_in[1];
    const float* W     = (const float*)d_in[2];
    float*       out   = (float*)d_out;
    int n = in_sizes[0];
    int V = in_sizes[2] / (DDIM * KDIM);   // 128
    int* ws = (int*)d_ws;

    ve_zero_ws<<<1, 384, 0, stream>>>(ws);
    ve_count<<<(n + 1023) / 1024, 1024, 0, stream>>>(vids, ws, n);
    ve_scan<<<1, 32, 0, stream>>>(ws, V);
    ve_scatter<<<(n + 1023) / 1024, 1024, 0, stream>>>(vids, ws, n);

    dim3 grid(V, 40);   // 5120 blocks * 8 waves, grid-striding over per-video tiles
    ve_wmma<<<grid, 256, 0, stream>>>(times, W, ws, out);
}